// PDECentralDiff_84722524880951
// MI455X (gfx1250) — compile-verified
//
#include <hip/hip_runtime.h>
#include <stdint.h>

typedef float f4 __attribute__((ext_vector_type(4)));

// Tile geometry: each block computes TD x TH x 8 floats of output.
#define TD 32
#define TH 32
#define HSLOTS ((TD + 2) * (TH + 2) * 2)   // halo tile in f4 slots = 2312 (36,992 B)
#define OUTS_PER_BLOCK (TD * TH * 2)       // 2048 f4 outputs per block

__global__ __launch_bounds__(256) void pde_central_diff_kernel(
    const float* __restrict__ p,   // [16,256,256,8] floats
    const f4* __restrict__ mu,     // minus_u, f4 view
    const f4* __restrict__ mv,     // minus_v, f4 view
    f4* __restrict__ out)          // trend_p, f4 view
{
    extern __shared__ f4 tile[];   // (TD+2) x (TH+2) x 2 f4

    const int tid = threadIdx.x;
    const int h0  = blockIdx.x * TH;
    const int d0  = blockIdx.y * TD;
    const int n   = blockIdx.z;

    // Dynamic LDS starts right after static LDS (0 here, but query to be robust).
    const uint32_t lds_base = (uint32_t)__builtin_amdgcn_groupstaticsize();

    // ---- Stage p halo tile into LDS via CDNA5 async global->LDS DMA ----
    // Slot s -> (ld in [0,TD+2), lh in [0,TH+2), lq in [0,2)); clamped neighbor
    // indices implement the reference's replicate padding.
    for (int s = tid; s < HSLOTS; s += 256) {
        const int lq = s & 1;
        const int t  = s >> 1;
        const int lh = t % (TH + 2);
        const int ld = t / (TH + 2);
        int gd = d0 + ld - 1; gd = gd < 0 ? 0 : (gd > 255 ? 255 : gd);
        int gh = h0 + lh - 1; gh = gh < 0 ? 0 : (gh > 255 ? 255 : gh);
        // byte offset of the f4 in p (max ~33.5 MB, fits signed 32-bit)
        const uint32_t goff = (uint32_t)(((((n * 256 + gd) * 256) + gh) * 2 + lq) * 16);
        const uint32_t loff = lds_base + (uint32_t)s * 16u;
        // GVS form: vdst = LDS byte address, vaddr = 32-bit offset, saddr = base
        asm volatile("global_load_async_to_lds_b128 %0, %1, %2"
                     :
                     : "v"(loff), "v"(goff), "s"(p)
                     : "memory");
    }
    // Wait for this wave's async copies, then make LDS visible workgroup-wide.
    asm volatile("s_wait_asynccnt 0" ::: "memory");
    __syncthreads();

    // ---- Compute: 8 f4 outputs per thread, fully coalesced ----
    // 1/(2*DX) = 1/(2*DY) = 128
    #pragma unroll
    for (int k = 0; k < OUTS_PER_BLOCK / 256; ++k) {
        const int o  = tid + k * 256;          // 0..2047
        const int oq = o & 1;                  // f4 half of the W row
        const int u  = o >> 1;                 // 0..1023
        const int oh = u & (TH - 1);           // 0..31
        const int od = u >> 5;                 // 0..31

        const f4 pdm = tile[((od    ) * (TH + 2) + (oh + 1)) * 2 + oq]; // p[d-1]
        const f4 pdp = tile[((od + 2) * (TH + 2) + (oh + 1)) * 2 + oq]; // p[d+1]
        const f4 phm = tile[((od + 1) * (TH + 2) + (oh    )) * 2 + oq]; // p[h-1]
        const f4 php = tile[((od + 1) * (TH + 2) + (oh + 2)) * 2 + oq]; // p[h+1]

        const int gidx = (((n * 256 + (d0 + od)) * 256) + (h0 + oh)) * 2 + oq;
        // One-touch streams: non-temporal so L2 capacity stays with p halos.
        const f4 a = __builtin_nontemporal_load(&mu[gidx]);
        const f4 b = __builtin_nontemporal_load(&mv[gidx]);

        f4 r;
        r.x = 128.0f * fmaf(a.x, pdp.x - pdm.x, b.x * (php.x - phm.x));
        r.y = 128.0f * fmaf(a.y, pdp.y - pdm.y, b.y * (php.y - phm.y));
        r.z = 128.0f * fmaf(a.z, pdp.z - pdm.z, b.z * (php.z - phm.z));
        r.w = 128.0f * fmaf(a.w, pdp.w - pdm.w, b.w * (php.w - phm.w));
        __builtin_nontemporal_store(r, &out[gidx]);
    }
}

extern "C" void kernel_launch(void* const* d_in, const int* in_sizes, int n_in,
                              void* d_out, int out_size, void* d_ws, size_t ws_size,
                              hipStream_t stream) {
    (void)in_sizes; (void)n_in; (void)d_ws; (void)ws_size; (void)out_size;

    // d_in[0] is the stacked [3,16,1,256,256,8] f32 tensor: p, minus_u, minus_v.
    const float* base = (const float*)d_in[0];
    const size_t plane = (size_t)16 * 256 * 256 * 8;   // 8,388,608 floats
    const float* p  = base;
    const f4*    mu = (const f4*)(base + plane);
    const f4*    mv = (const f4*)(base + 2 * plane);
    f4* outv = (f4*)d_out;

    dim3 grid(256 / TH, 256 / TD, 16);   // (8, 8, 16) blocks
    dim3 block(256);                     // 8 wave32 waves per block
    const size_t shmem = (size_t)HSLOTS * sizeof(f4);  // 36,992 bytes

    pde_central_diff_kernel<<<grid, block, shmem, stream>>>(p, mu, mv, outv);
}